// dcnv4_24790551232949
// MI455X (gfx1250) — compile-verified
//
#include <hip/hip_runtime.h>

// ---- problem constants (from the reference) ----
#define CCH   256      // channels
#define GRP   16       // groups
#define GC    16       // channels per group
#define PTS   9        // kernel taps (3x3)
#define OMC   432      // G*P*3 offset/mask channels
#define HH    64
#define WW    64
#define LPIX  4096     // H*W
#define NBATCH 8

typedef float v2f __attribute__((ext_vector_type(2)));
typedef float v8f __attribute__((ext_vector_type(8)));

// ============================================================================
// GEMM: Y[M,Nc] = X[M,K=256] @ Wt[Nc,K]^T + bias[Nc]
// fp32 via V_WMMA_F32_16X16X4_F32 (wave32).
//
// Block = 128 threads = 4 waves laid out 2(M) x 2(N).
// Wave tile = 32M x 64N: 2 A fragments x 4 B fragments -> 8 WMMAs per k-step
// over 8 independent accumulator chains (hides WMMA latency), with only
// 6 global_load_b64 per 8 WMMAs. Block tile = 64M x 128N.
//
// Fragment layout per CDNA5 ISA (32-bit A 16x4 / B 4x16):
//   lanes 0-15 : elements K = k+0, k+1   (2 VGPRs)
//   lanes 16-31: elements K = k+2, k+3
// -> each lane loads a contiguous float2 at (row, k + 2*half).
// C/D 16x16 f32: lanes 0-15 hold tile rows 0..7, lanes 16-31 rows 8..15,
// N = lane & 15.
//
// OOB columns (Nc=432 tail): B loads are clamped to row Nc-1 (keeps EXEC
// uniform around WMMA); garbage in B column n only touches D column n,
// whose lanes simply skip the store. No masking VALU needed.
// ============================================================================
__global__ __launch_bounds__(128) void gemm_wmma_f32(
    const float* __restrict__ X, const float* __restrict__ Wt,
    const float* __restrict__ bias, float* __restrict__ Y,
    int M, int Nc)
{
    const int K    = CCH;
    const int tid  = threadIdx.x;
    const int wave = tid >> 5;
    const int lane = tid & 31;
    const int half = lane >> 4;     // selects K pair within the 4-step
    const int lm   = lane & 15;     // M index (A) / N index (B)

    const int mw = wave >> 1;       // wave's M sub-tile (0..1)
    const int nw = wave & 1;        // wave's N sub-tile (0..1)

    const int m_base = blockIdx.x * 64 + mw * 32;
    const int n_base = blockIdx.y * 128 + nw * 64;

    // 32-bit element offsets (all tensors < 2^32 bytes) -> saddr+voffset loads
    const unsigned aoff0 = (unsigned)(m_base + lm) * K + half * 2;
    const unsigned aoff1 = aoff0 + 16u * K;

    unsigned boff[4];
    int      col[4];
    bool     ok[4];
    #pragma unroll
    for (int j = 0; j < 4; ++j) {
        col[j] = n_base + 16 * j + lm;
        ok[j]  = col[j] < Nc;
        const int cc = ok[j] ? col[j] : (Nc - 1);   // clamp, keep loads uniform
        boff[j] = (unsigned)cc * K + half * 2;
    }

    v8f acc[2][4] = {};

    #pragma unroll 4
    for (int k = 0; k < K; k += 4) {
        const v2f a0 = *(const v2f*)&X[aoff0 + k];
        const v2f a1 = *(const v2f*)&X[aoff1 + k];
        v2f b[4];
        #pragma unroll
        for (int j = 0; j < 4; ++j)
            b[j] = *(const v2f*)&Wt[boff[j] + k];

        #pragma unroll
        for (int j = 0; j < 4; ++j) {
            acc[0][j] = __builtin_amdgcn_wmma_f32_16x16x4_f32(
                            false, a0, false, b[j], (short)0, acc[0][j],
                            false, false);
            acc[1][j] = __builtin_amdgcn_wmma_f32_16x16x4_f32(
                            false, a1, false, b[j], (short)0, acc[1][j],
                            false, false);
        }
    }

    #pragma unroll
    for (int j = 0; j < 4; ++j) {
        if (!ok[j]) continue;
        const float bb = bias[col[j]];
        #pragma unroll
        for (int i = 0; i < 2; ++i) {
            const int rbase = m_base + i * 16 + half * 8;
            #pragma unroll
            for (int r = 0; r < 8; ++r)
                Y[(size_t)(rbase + r) * Nc + col[j]] = acc[i][j][r] + bb;
        }
    }
}

// ============================================================================
// Deformable bilinear gather + mask aggregation.
// Padded-image coords reduce exactly to unpadded sampling:
//   px = x + (p/3 - 1) + off_x,  py = y + (p%3 - 1) + off_y, zero outside.
// Block = 256 threads = 4 pixels x 16 groups x 4 lanes; each lane owns a
// float4 (4 channels). om for the 4 pixels is staged through LDS.
// ============================================================================
__global__ __launch_bounds__(256) void dcn_gather(
    const float* __restrict__ v, const float* __restrict__ om,
    float* __restrict__ agg)
{
    __shared__ float som[4 * OMC];

    const int tid = threadIdx.x;
    const int pg0 = blockIdx.x * 4;

    for (int i = tid; i < 4 * OMC; i += 256)
        som[i] = om[(size_t)pg0 * OMC + i];
    __syncthreads();

    const int psub = tid >> 6;         // pixel within block: 0..3
    const int g    = (tid >> 2) & 15;  // group
    const int c4   = (tid & 3) * 4;    // first channel of this lane's float4

    const int pg  = pg0 + psub;
    const int n   = pg >> 12;          // / 4096
    const int rem = pg & 4095;
    const int yy  = rem >> 6;
    const int xx  = rem & 63;

    const float* __restrict__ omg   = som + psub * OMC + g * 27;
    const float* __restrict__ vbase = v + (size_t)n * LPIX * CCH + g * GC + c4;

    float4 acc = make_float4(0.f, 0.f, 0.f, 0.f);

    #pragma unroll
    for (int p = 0; p < PTS; ++p) {
        const float px = (float)xx + (float)(p / 3 - 1) + omg[2 * p + 0];
        const float py = (float)yy + (float)(p % 3 - 1) + omg[2 * p + 1];
        const float m  = omg[2 * PTS + p];

        const float x0f = floorf(px);
        const float y0f = floorf(py);
        const float fx  = px - x0f;
        const float fy  = py - y0f;
        const int   x0  = (int)x0f;
        const int   y0  = (int)y0f;

        const float wq[4] = { (1.f - fx) * (1.f - fy), fx * (1.f - fy),
                              (1.f - fx) * fy,         fx * fy };
        #pragma unroll
        for (int q = 0; q < 4; ++q) {
            const int xi = x0 + (q & 1);
            const int yi = y0 + (q >> 1);
            if (xi >= 0 && xi < WW && yi >= 0 && yi < HH) {
                const float4 s =
                    *(const float4*)(vbase + (size_t)(yi * WW + xi) * CCH);
                const float wm = wq[q] * m;
                acc.x += wm * s.x; acc.y += wm * s.y;
                acc.z += wm * s.z; acc.w += wm * s.w;
            }
        }
    }

    float* op = agg + (size_t)pg * CCH + g * GC + c4;
    *(float4*)op = acc;
}

// ============================================================================
// Host-side launch: v-GEMM, om-GEMM, gather, out-GEMM  (all on `stream`)
// ============================================================================
extern "C" void kernel_launch(void* const* d_in, const int* in_sizes, int n_in,
                              void* d_out, int out_size, void* d_ws, size_t ws_size,
                              hipStream_t stream)
{
    const float* x       = (const float*)d_in[0];
    const float* value_w = (const float*)d_in[1];
    const float* value_b = (const float*)d_in[2];
    const float* om_w    = (const float*)d_in[3];
    const float* om_b    = (const float*)d_in[4];
    const float* out_w   = (const float*)d_in[5];
    const float* out_b   = (const float*)d_in[6];
    float*       out     = (float*)d_out;

    const int M = NBATCH * LPIX;          // 32768 rows

    float* ws   = (float*)d_ws;
    float* vbuf = ws;                                  // M x 256
    float* ombf = vbuf + (size_t)M * CCH;              // M x 432
    float* abuf = ombf + (size_t)M * OMC;              // M x 256

    const dim3 blk(128);

    // v = x @ value_w^T + value_b
    gemm_wmma_f32<<<dim3(M / 64, (CCH + 127) / 128), blk, 0, stream>>>(
        x, value_w, value_b, vbuf, M, CCH);

    // om = x @ om_w^T + om_b
    gemm_wmma_f32<<<dim3(M / 64, (OMC + 127) / 128), blk, 0, stream>>>(
        x, om_w, om_b, ombf, M, OMC);

    // deformable gather + mask aggregation
    dcn_gather<<<dim3(M / 4), dim3(256), 0, stream>>>(vbuf, ombf, abuf);

    // out = agg @ out_w^T + out_b
    gemm_wmma_f32<<<dim3(M / 64, (CCH + 127) / 128), blk, 0, stream>>>(
        abuf, out_w, out_b, out, M, CCH);
}